// DelayLayer_50362786513382
// MI455X (gfx1250) — compile-verified
//
#include <hip/hip_runtime.h>

// Delay-and-sum beamformer for MI455X (gfx1250).
//
// Roofline: output stores = 128 MB -> ~5.5us floor @ 23.3 TB/s; input is
// 1.1 MB (L2-resident). No GEMM structure -> WMMA inapplicable; the gfx1250
// features used: async global->LDS staging (global_load_async_to_lds_b128 +
// s_wait_asynccnt) so the 2 gathers/pair run on the DS pipe while VMEM
// streams the 128 MB of output stores.
//
// Branchless inner loop, ~15 VALU/pair: idx = (dist/C + T0)/T_DT folded to
// one fma (T0/T_DT == -932.0 exactly); gate predicate reused for both the
// idx->0 and y0->0 selects (gated => i0==0 && wb==0, matching the
// reference's zeroed first time-sample). ~6us VALU vs 5.5us store floor.

#define TPB    256
#define CHUNK  4          // sensors staged in LDS per pass (4 * 2168 * 4B = 34.7 KB)
#define T_LEN  2168
#define NXY    262144     // 512*512
#define PPT    8          // pixels per thread (two float4 stores per channel)

struct SM {
  float sig[CHUNK][T_LEN]; // staged signals (filled by async loads)
  float sx[32];            // sensor x for this group
  float sy[32];            // sensor y for this group
};

__global__ __launch_bounds__(TPB) void das_kernel(const float* __restrict__ x,
                                                  float* __restrict__ out) {
  __shared__ SM sm;
  const int tid = threadIdx.x;
  const int g   = blockIdx.y;          // sensor group 0..3

  // Sensor ring positions: phi = linspace(0, 2*PI, 128), r=0.11, ang=PI
  if (tid < 32) {
    const float phi = (float)(g * 32 + tid) * (6.283184f / 127.0f);
    sm.sx[tid] = 0.11f * cosf(phi + 3.141592f);
    sm.sy[tid] = 0.11f * sinf(phi + 3.141592f);
  }

  // Each thread: PPT consecutive iy pixels of one ix row.
  const int p    = blockIdx.x * (TPB * PPT) + tid * PPT;
  const int ix   = p >> 9;
  const int iy0  = p & 511;
  const float step = 512.0f / 511.0f;      // linspace(-256,256,512) step
  const float gsc  = 0.15f / 512.0f;       // G_DX == G_DY
  const float gxv  = (-256.0f + (float)ix * step) * gsc;
  float gy[PPT];
#pragma unroll
  for (int k = 0; k < PPT; ++k)
    gy[k] = (-256.0f + (float)(iy0 + k) * step) * gsc;

  float acc[PPT];
#pragma unroll
  for (int k = 0; k < PPT; ++k) acc[k] = 0.0f;

  // idx = dist * 1/(C*T_DT) + T0/T_DT  (T0/T_DT = -932.0 exactly)
  const float K_IDX = 1.0f / (1500.0f * 2.5e-08f);
  const float B_IDX = -932.0f;

  // Low 32 bits of a generic LDS pointer == workgroup-relative LDS byte
  // offset (aperture form), which is exactly what the async-load VDST wants.
  const unsigned lds_base = (unsigned)(unsigned long long)(const void*)&sm.sig[0][0];

  for (int cb = 0; cb < 32; cb += CHUNK) {
    __syncthreads();  // prev chunk consumed; sensor pos ready on iter 0

    // CHUNK consecutive sensors are contiguous in x: one linear 34,688 B copy.
    const float4* src = (const float4*)(x + (size_t)(g * 32 + cb) * (size_t)T_LEN);
    for (int i = tid; i < (CHUNK * T_LEN) / 4; i += TPB) {
      const unsigned dst = lds_base + (unsigned)(i * 16);
      const unsigned long long ga = (unsigned long long)(const void*)(src + i);
      asm volatile("global_load_async_to_lds_b128 %0, %1, off"
                   :: "v"(dst), "v"(ga) : "memory");
    }
    asm volatile("s_wait_asynccnt 0" ::: "memory");  // this wave's copies landed
    __syncthreads();                                  // all waves' copies landed

#pragma unroll
    for (int c = 0; c < CHUNK; ++c) {
      const float sx  = sm.sx[cb + c];
      const float sy  = sm.sy[cb + c];
      const float dxv = gxv - sx;
      const float dx2 = dxv * dxv;
#pragma unroll
      for (int k = 0; k < PPT; ++k) {
        const float dyv  = gy[k] - sy;
        const float dist = sqrtf(dx2 + dyv * dyv);
        float idx = fmaf(dist, K_IDX, B_IDX);
        const bool gate = (idx > 2166.0f) || (idx < 200.0f);
        idx = gate ? 0.0f : idx;
        const float d0 = floorf(idx);
        const int   i0 = (int)d0;                 // in [0, 2166]
        const float wb = idx - d0;
        const float wa = 1.0f - wb;               // == (d0+1)-idx exactly here
        float y0       = sm.sig[c][i0];
        const float y1 = sm.sig[c][i0 + 1];
        y0 = gate ? 0.0f : y0;                    // forward zeroes t=0 sample
        acc[k] = fmaf(wa, y0, fmaf(wb, y1, acc[k]));
      }
    }
  }

  // Broadcast the group sum to all 32 channels of this group (float4 stores).
  const float4 v0 = make_float4(acc[0], acc[1], acc[2], acc[3]);
  const float4 v1 = make_float4(acc[4], acc[5], acc[6], acc[7]);
  float* obase = out + (size_t)g * 32 * NXY + (size_t)ix * 512 + iy0;
#pragma unroll
  for (int cc = 0; cc < 32; ++cc) {
    float4* q = reinterpret_cast<float4*>(obase + (size_t)cc * NXY);
    q[0] = v0;
    q[1] = v1;
  }
}

extern "C" void kernel_launch(void* const* d_in, const int* in_sizes, int n_in,
                              void* d_out, int out_size, void* d_ws, size_t ws_size,
                              hipStream_t stream) {
  (void)in_sizes; (void)n_in; (void)d_ws; (void)ws_size; (void)out_size;
  const float* x = (const float*)d_in[0];   // [1, 128, 2168] f32
  float* out     = (float*)d_out;           // [1, 128, 512, 512] f32
  dim3 grid(NXY / (TPB * PPT), 4);          // (128 pixel-blocks, 4 groups)
  das_kernel<<<grid, TPB, 0, stream>>>(x, out);
}